// ClassificationHead_14345190769322
// MI455X (gfx1250) — compile-verified
//
#include <hip/hip_runtime.h>
#include <hip/hip_bf16.h>
#include <stdint.h>

typedef __attribute__((ext_vector_type(2))) float v2f;
typedef __attribute__((ext_vector_type(8))) float v8f;
typedef __attribute__((ext_vector_type(4))) uint32_t u32x4;
typedef __attribute__((ext_vector_type(8))) uint32_t u32x8;

#define NTASK   512
#define NQ      75
#define NS      25
#define DIM     640
#define NW      5
#define RLAMBDA 50.0f
#define KC      64

// LDS strides (floats). Both produced by TDM pad insertion: pad every 64 DWORDs.
// support: +2 DWORDs per 64  -> row stride 660  (660 % 64 == 20: 16 strided rows -> 16 banks)
// query:   +4 DWORDs per 64  -> row stride 68   (one 64-col chunk per row)
#define SUP_STRIDE 660
#define G_STRIDE   33
#define WDT_STRIDE 648
#define Q_STRIDE   68

// LDS layout (floats)
#define OFF_SUP   0
#define OFF_G     (32 * SUP_STRIDE)              // 21120
#define OFF_PIV   (OFF_G + 32 * G_STRIDE)        // 22176
#define OFF_ALPHA (OFF_PIV + 32)                 // 22208
#define OFF_WDT   (OFF_ALPHA + 26 * 8)           // 22416
#define SMEM_FLOATS (OFF_WDT + 16 * WDT_STRIDE)  // 32784 floats = 131136 B
// query double-buffers alias the (dead after phase D) support region:
#define OFF_Q0    0
#define OFF_Q1    (80 * Q_STRIDE)                // 5440 (both fit inside 21120)

// ---- Tensor Data Mover: 2D tile load, D# built per CDNA5 ISA ----
// 2-group form (2D tensor): tensor_load_to_lds s[g0 x4], s[g1 x8]
__device__ __forceinline__ void tdm_load_2d(uint32_t lds_byte_addr, const void* gptr,
                                            uint32_t dim0, uint32_t dim1,
                                            uint32_t row_stride_elems,
                                            uint32_t pad_amount_enc /* 2^? DWORDs-1 enc */)
{
    uint64_t ga = (uint64_t)(uintptr_t)gptr;
    u32x4 g0;
    g0[0] = 1u;                                   // count=1 valid user descriptor
    g0[1] = lds_byte_addr;                        // lds_addr
    g0[2] = (uint32_t)(ga & 0xffffffffu);         // global_addr[31:0]
    g0[3] = (uint32_t)((ga >> 32) & 0x01ffffffu)  // global_addr[56:32]
            | (2u << 30);                         // type = 2 ("image")
    u32x8 g1;
    g1[0] = (2u << 16)                            // data_size = 4 bytes
          | (1u << 20)                            // pad_enable
          | (5u << 22)                            // pad_interval: 64 DWORDs
          | (pad_amount_enc << 25);               // pad_amount (enc: 1 -> 2 DW, 3 -> 4 DW)
    g1[1] = (dim0 & 0xffffu) << 16;               // tensor_dim0[15:0] (abar addr = 0)
    g1[2] = (dim0 >> 16) | ((dim1 & 0xffffu) << 16);  // tensor_dim0[31:16] | tensor_dim1[15:0]
    g1[3] = (dim1 >> 16) | (dim0 << 16);          // tensor_dim1[31:16] | tile_dim0 (=dim0, <=640)
    g1[4] = dim1 & 0xffffu;                       // tile_dim1 (=dim1) | tile_dim2 = 0
    g1[5] = row_stride_elems;                     // tensor_dim0_stride[31:0]
    g1[6] = 0u;                                   // stride[47:32] | tensor_dim1_stride[15:0]
    g1[7] = 0u;
    asm volatile("tensor_load_to_lds %0, %1" :: "s"(g0), "s"(g1) : "memory");
}

__global__ __launch_bounds__(256) void r2d2_head_kernel(
    const float* __restrict__ query,     // (T, NQ, DIM)
    const float* __restrict__ support,   // (T, NS, DIM)
    const int*   __restrict__ labels,    // (T, NS)
    const float* __restrict__ scale,     // (1,)
    float*       __restrict__ out)       // (T, NQ, NW)
{
    extern __shared__ float smem[];
    float* s_sup   = smem + OFF_SUP;
    float* s_G     = smem + OFF_G;
    float* s_piv   = smem + OFF_PIV;
    float* s_alpha = smem + OFF_ALPHA;
    float* s_wdt   = smem + OFF_WDT;

    const int task = blockIdx.x;
    const int tid  = threadIdx.x;
    const int wv   = tid >> 5;        // wave id (wave32)
    const int lane = tid & 31;
    const int hi   = lane >> 4;       // lane half -> K offset 0 or 2
    const int l16  = lane & 15;

    const uint32_t lds_base = (uint32_t)(uintptr_t)smem;  // low 32 bits of flat LDS ptr = DS addr
    const float* supg = support + (size_t)task * NS * DIM;
    const float* qg   = query   + (size_t)task * NQ * DIM;

    // ---- Phase A: TDM DMA the support tile (25x640, padded to stride 660) into LDS ----
    if (wv == 0) {
        tdm_load_2d(lds_base + OFF_SUP * 4u, supg, DIM, NS, DIM, /*pad 2 DW*/1u);
    }
    // meanwhile: zero the support pad rows (25..31) and WdT pad rows (5..15)
    for (int i = tid; i < 7 * SUP_STRIDE; i += 256)  s_sup[25 * SUP_STRIDE + i] = 0.0f;
    for (int i = tid; i < 11 * WDT_STRIDE; i += 256) s_wdt[5 * WDT_STRIDE + i]  = 0.0f;
    if (wv == 0) __builtin_amdgcn_s_wait_tensorcnt(0);
    __syncthreads();

    // ---- Phase B: G = S*S^T + lambda*I (padded 32x32) via fp32 WMMA 16x16x4 ----
    if (wv < 4) {
        const int ti = wv >> 1, tj = wv & 1;
        const float* arow = &s_sup[(ti * 16 + l16) * SUP_STRIDE + 2 * hi];
        const float* brow = &s_sup[(tj * 16 + l16) * SUP_STRIDE + 2 * hi];
        v8f acc = {};
        for (int kc = 0; kc < DIM / KC; ++kc) {
            const float* ap = arow + kc * 66;   // 64 data + 2 pad DWORDs per chunk
            const float* bp = brow + kc * 66;
#pragma unroll 4
            for (int k = 0; k < KC; k += 4) {
                v2f a = *(const v2f*)(ap + k);
                v2f b = *(const v2f*)(bp + k);
                acc = __builtin_amdgcn_wmma_f32_16x16x4_f32(
                    false, a, false, b, (short)0, acc, false, false);
            }
        }
#pragma unroll
        for (int v = 0; v < 8; ++v) {
            int M = ti * 16 + v + 8 * hi;
            int N = tj * 16 + l16;
            float g = acc[v];
            if (M == N) g += RLAMBDA;
            s_G[M * G_STRIDE + N] = g;
        }
    }
    __syncthreads();

    // ---- Phase C: Gauss-Jordan solve (G | one_hot) -> alpha (25x5); SPD+ridge, no pivoting ----
    float row[NS + NW];
    if (tid < NS) {
#pragma unroll
        for (int j = 0; j < NS; ++j) row[j] = s_G[tid * G_STRIDE + j];
        const int lab = labels[task * NS + tid];
#pragma unroll
        for (int w = 0; w < NW; ++w) row[NS + w] = (lab == w) ? 1.0f : 0.0f;
    }
    __syncthreads();
    for (int p = 0; p < NS; ++p) {
        if (tid == p) {
            float inv = 1.0f / row[p];
#pragma unroll
            for (int j = 0; j < NS + NW; ++j) { row[j] *= inv; s_piv[j] = row[j]; }
        }
        __syncthreads();
        if (tid < NS && tid != p) {
            float f = row[p];
#pragma unroll
            for (int j = 0; j < NS + NW; ++j) row[j] = fmaf(-f, s_piv[j], row[j]);
        }
        __syncthreads();
    }
    if (tid < NS) {
#pragma unroll
        for (int w = 0; w < NW; ++w) s_alpha[tid * 8 + w] = row[NS + w];
    }
    __syncthreads();

    // ---- Phase D: W_dual^T = (S^T alpha)^T stored (NW x DIM, stride 648); rows 5..15 stay 0 ----
    for (int d = tid; d < DIM; d += 256) {
        const int dp = d + 2 * (d >> 6);          // physical column in padded support
        float a0 = 0.f, a1 = 0.f, a2 = 0.f, a3 = 0.f, a4 = 0.f;
#pragma unroll
        for (int s = 0; s < NS; ++s) {
            float sv = s_sup[s * SUP_STRIDE + dp];
            a0 = fmaf(sv, s_alpha[s * 8 + 0], a0);
            a1 = fmaf(sv, s_alpha[s * 8 + 1], a1);
            a2 = fmaf(sv, s_alpha[s * 8 + 2], a2);
            a3 = fmaf(sv, s_alpha[s * 8 + 3], a3);
            a4 = fmaf(sv, s_alpha[s * 8 + 4], a4);
        }
        s_wdt[0 * WDT_STRIDE + d] = a0;
        s_wdt[1 * WDT_STRIDE + d] = a1;
        s_wdt[2 * WDT_STRIDE + d] = a2;
        s_wdt[3 * WDT_STRIDE + d] = a3;
        s_wdt[4 * WDT_STRIDE + d] = a4;
    }
    __syncthreads();   // support region is now dead -> query buffers may alias it

    // ---- Phase E: logits = Q * W_dual, query streamed by double-buffered TDM DMA ----
    float* s_qb[2] = { smem + OFF_Q0, smem + OFF_Q1 };
    // zero query pad rows (75..79) of both buffers; disjoint from TDM target rows 0..74
    for (int i = tid; i < 5 * Q_STRIDE; i += 256) {
        s_qb[0][75 * Q_STRIDE + i] = 0.0f;
        s_qb[1][75 * Q_STRIDE + i] = 0.0f;
    }
    if (wv == 0) {   // kick off chunk 0
        tdm_load_2d(lds_base + OFF_Q0 * 4u, qg, KC, NQ, DIM, /*pad 4 DW*/3u);
    }

    v8f lacc = {};
    const int NCHUNK = DIM / KC;       // 10
    for (int i = 0; i < NCHUNK; ++i) {
        if (wv == 0) {
            if (i + 1 < NCHUNK) {       // prefetch next chunk into the other buffer
                tdm_load_2d(lds_base + (((i + 1) & 1) ? OFF_Q1 : OFF_Q0) * 4u,
                            qg + (i + 1) * KC, KC, NQ, DIM, 3u);
                __builtin_amdgcn_s_wait_tensorcnt(1);   // chunk i has landed
            } else {
                __builtin_amdgcn_s_wait_tensorcnt(0);
            }
        }
        __syncthreads();               // chunk i visible to all waves
        if (wv < 5) {
            const float* arow = &s_qb[i & 1][(wv * 16 + l16) * Q_STRIDE + 2 * hi];
            const float* brow = &s_wdt[l16 * WDT_STRIDE + i * KC + 2 * hi];
#pragma unroll 4
            for (int k = 0; k < KC; k += 4) {
                v2f a = *(const v2f*)(arow + k);
                v2f b = *(const v2f*)(brow + k);
                lacc = __builtin_amdgcn_wmma_f32_16x16x4_f32(
                    false, a, false, b, (short)0, lacc, false, false);
            }
        }
        __syncthreads();               // readers done before this buffer is re-DMAed
    }

    // ---- scale + store (T, NQ, NW) ----
    if (wv < 5) {
        const float scl = scale[0];
#pragma unroll
        for (int v = 0; v < 8; ++v) {
            int q = wv * 16 + v + 8 * hi;
            int w = l16;
            if (q < NQ && w < NW)
                out[((size_t)task * NQ + q) * NW + w] = scl * lacc[v];
        }
    }
}

extern "C" void kernel_launch(void* const* d_in, const int* in_sizes, int n_in,
                              void* d_out, int out_size, void* d_ws, size_t ws_size,
                              hipStream_t stream) {
    const float* query   = (const float*)d_in[0];
    const float* support = (const float*)d_in[1];
    const int*   labels  = (const int*)d_in[2];
    const float* scale   = (const float*)d_in[5];
    float* out = (float*)d_out;
    (void)in_sizes; (void)n_in; (void)d_ws; (void)ws_size; (void)out_size;

    dim3 grid(NTASK), block(256);
    size_t smem_bytes = (size_t)SMEM_FLOATS * sizeof(float);
    r2d2_head_kernel<<<grid, block, smem_bytes, stream>>>(query, support, labels, scale, out);
}